// GNNVariationalExample_27925877358777
// MI455X (gfx1250) — compile-verified
//
#include <hip/hip_runtime.h>
#include <math.h>

typedef __attribute__((ext_vector_type(2))) float v2f;
typedef __attribute__((ext_vector_type(8))) float v8f;
typedef __attribute__((ext_vector_type(4))) int   v4i;
typedef __attribute__((address_space(1))) v4i g_v4i;
typedef __attribute__((address_space(3))) v4i l_v4i;

#define B_GR   64
#define N_NODE 1000
#define F_DIM  64
#define E_NUM  32000
#define IN_DIM 64000          // N_NODE * F_DIM
#define HID    512
#define LAT    128
#define ROWS   (B_GR * N_NODE) // 64000 flattened (batch,node) rows

// ---- workspace layout (float offsets) ----
#define WS_S    0L
#define WS_XS   (WS_S  + (long)ROWS * F_DIM)   // 4,096,000
#define WS_DEG  (WS_XS + (long)ROWS * F_DIM)   // 8,192,000
#define WS_PRE  (WS_DEG + 1024)                // 8,193,024
#define WS_HE   (WS_PRE + (long)B_GR * HID)    // 8,225,792  (zero region ends here)
#define WS_Z    (WS_HE  + (long)B_GR * HID)
#define WS_HD   (WS_Z   + (long)B_GR * LAT)

// -------- CDNA5 async-to-LDS availability --------
#if defined(__has_builtin)
#  if __has_builtin(__builtin_amdgcn_global_load_async_to_lds_b128) && \
      __has_builtin(__builtin_amdgcn_s_wait_asynccnt)
#    define HAVE_ASYNC_LDS 1
#  endif
#endif
#ifndef HAVE_ASYNC_LDS
#  define HAVE_ASYNC_LDS 0
#endif

#if HAVE_ASYNC_LDS
#  define ASYNC_WAIT(n) do { __builtin_amdgcn_s_wait_asynccnt(n); \
                             __asm__ volatile("" ::: "memory"); } while (0)
#else
#  define ASYNC_WAIT(n) do { __asm__ volatile("" ::: "memory"); } while (0)
#endif

// ---------------- WMMA helpers (V_WMMA_F32_16X16X4_F32) ----------------
// A tile 16x4 f32: lanes 0-15 hold M=lane, VGPR0=K0,VGPR1=K1; lanes 16-31 hold K2,K3
__device__ __forceinline__ v2f load_a_f32(const float* __restrict__ A, int lda,
                                          int r0, int k0, int lane) {
    int m  = lane & 15;
    int kk = (lane >> 4) << 1;           // 0 or 2
    const float* p = A + (long)(r0 + m) * lda + k0 + kk;
    v2f a; a.x = p[0]; a.y = p[1];
    return a;
}
// B tile 4x16 f32 from row-major weight [K,N], ldb = N (direct global path)
__device__ __forceinline__ v2f load_b_f32(const float* __restrict__ Bm, long ldb,
                                          int k0, long n0, int lane) {
    int n  = lane & 15;
    int kk = (lane >> 4) << 1;
    v2f b;
    b.x = Bm[(long)(k0 + kk)     * ldb + n0 + n];
    b.y = Bm[(long)(k0 + kk + 1) * ldb + n0 + n];
    return b;
}
// B tile 4x16 f32 from LDS chunk laid out [32 rows][16 cols]
__device__ __forceinline__ v2f load_b_lds(const float* lbuf, int ks, int lane) {
    int n  = lane & 15;
    int kk = (lane >> 4) << 1;
    v2f b;
    b.x = lbuf[(ks + kk)     * 16 + n];
    b.y = lbuf[(ks + kk + 1) * 16 + n];
    return b;
}
__device__ __forceinline__ v8f wmma4(v2f a, v2f b, v8f c) {
    return __builtin_amdgcn_wmma_f32_16x16x4_f32(false, a, false, b, (short)0, c,
                                                 false, false);
}
#define V8F_ZERO {0.f,0.f,0.f,0.f,0.f,0.f,0.f,0.f}

// Stage a 32-row x 16-col fp32 weight chunk (2 KB) into LDS.
// Global layout: row stride ldbF floats, 64 contiguous bytes per row.
// 128 x 16B segments; lane handles segments lane+32*i (4 async b128 issues).
__device__ __forceinline__ void stage_w(const float* __restrict__ gW, long ldbF,
                                        int k0, long n0, float* lbuf, int lane) {
#pragma unroll
    for (int i = 0; i < 4; ++i) {
        int sg   = lane + 32 * i;
        int row  = sg >> 2;
        int part = sg & 3;
        const float* g = gW + (long)(k0 + row) * ldbF + n0 + part * 4;
#if HAVE_ASYNC_LDS
        __builtin_amdgcn_global_load_async_to_lds_b128(
            (g_v4i*)g, (l_v4i*)(lbuf + sg * 4), 0, 0);
#else
        float4 v = *(const float4*)g;
        *(float4*)(lbuf + sg * 4) = v;
#endif
    }
}

// One 32-deep K chunk of WMMAs for 4 M-tiles (M = 64), B from LDS.
__device__ __forceinline__ void mma_chunk(v8f acc[4], const float* lbuf,
                                          const float* __restrict__ A, int lda,
                                          int kg0, int lane) {
#pragma unroll
    for (int ks = 0; ks < 32; ks += 4) {
        v2f b = load_b_lds(lbuf, ks, lane);
#pragma unroll
        for (int t = 0; t < 4; ++t) {
            v2f a = load_a_f32(A, lda, t * 16, kg0 + ks, lane);
            acc[t] = wmma4(a, b, acc[t]);
        }
    }
}

// ---------------- kernels ----------------
__global__ void zero_kernel(float* __restrict__ p, long n) {
    long i = (long)blockIdx.x * blockDim.x + threadIdx.x;
    if (i < n) p[i] = 0.0f;
}

__global__ void deg_kernel(const int* __restrict__ dst, float* __restrict__ deg) {
    int e = blockIdx.x * blockDim.x + threadIdx.x;
    if (e < E_NUM) atomicAdd(&deg[dst[e]], 1.0f);
}

// scatter-add raw source features into s[b, dst, :]  (x is L2-resident)
__global__ void scatter_kernel(const float* __restrict__ x, const int* __restrict__ src,
                               const int* __restrict__ dst, float* __restrict__ s) {
    int t = threadIdx.x;                       // 256 threads = 4 edges x 64 feats
    int e = blockIdx.x * 4 + (t >> 6);
    int f = t & 63;
    int b = blockIdx.y;
    float v = x[(long)b * IN_DIM + (long)src[e] * F_DIM + f];
    atomicAdd(&s[(long)b * IN_DIM + (long)dst[e] * F_DIM + f], v);
}

// xs = leaky( (s @ W_msg + deg*b_msg)/max(deg,1) + x ),  one wave per 16-row tile
__global__ void conv_kernel(const float* __restrict__ s, const float* __restrict__ x,
                            const float* __restrict__ deg, const float* __restrict__ Wmsg,
                            const float* __restrict__ bmsg, float* __restrict__ xs) {
    int lane = threadIdx.x;
    int r0 = blockIdx.x * 16;
    v8f acc[4] = {V8F_ZERO, V8F_ZERO, V8F_ZERO, V8F_ZERO};
    for (int k0 = 0; k0 < F_DIM; k0 += 4) {
        v2f a = load_a_f32(s, F_DIM, r0, k0, lane);
#pragma unroll
        for (int t = 0; t < 4; ++t) {
            v2f b = load_b_f32(Wmsg, F_DIM, k0, t * 16, lane);
            acc[t] = wmma4(a, b, acc[t]);
        }
    }
    int n_lo = lane & 15;
    int mo   = (lane >> 4) * 8;
#pragma unroll
    for (int t = 0; t < 4; ++t) {
        int col = t * 16 + n_lo;
        float bc = bmsg[col];
#pragma unroll
        for (int j = 0; j < 8; ++j) {
            int r = r0 + j + mo;
            float dg = deg[r % N_NODE];
            float v = (acc[t][j] + dg * bc) / fmaxf(dg, 1.0f) + x[(long)r * F_DIM + col];
            v = (v >= 0.f) ? v : 0.01f * v;
            xs[(long)r * F_DIM + col] = v;
        }
    }
}

// encoder big GEMM, K-split, async-LDS double-buffered weight streaming:
// grid = (HID/16 N-tiles, IN_DIM/512 K-slabs), 1 wave/block
__global__ void enc1_kernel(const float* __restrict__ xs, const float* __restrict__ We1,
                            float* __restrict__ preact) {
    __shared__ float lb[2][512];
    int lane  = threadIdx.x;
    int n0    = blockIdx.x * 16;
    int kbase = blockIdx.y * 512;
    v8f acc[4] = {V8F_ZERO, V8F_ZERO, V8F_ZERO, V8F_ZERO};

    stage_w(We1, HID, kbase, n0, lb[0], lane);
    for (int c = 0; c < 15; ++c) {
        stage_w(We1, HID, kbase + (c + 1) * 32, n0, lb[(c + 1) & 1], lane);
        ASYNC_WAIT(4);                     // chunk c landed; chunk c+1 in flight
        mma_chunk(acc, lb[c & 1], xs, IN_DIM, kbase + c * 32, lane);
    }
    ASYNC_WAIT(0);
    mma_chunk(acc, lb[1], xs, IN_DIM, kbase + 15 * 32, lane);

    int n_lo = lane & 15;
    int mo   = (lane >> 4) * 8;
#pragma unroll
    for (int t = 0; t < 4; ++t)
#pragma unroll
        for (int j = 0; j < 8; ++j) {
            int r = t * 16 + j + mo;
            atomicAdd(&preact[(long)r * HID + n0 + n_lo], acc[t][j]);
        }
}

__global__ void bias_relu_kernel(const float* __restrict__ in, const float* __restrict__ bias,
                                 float* __restrict__ out, int n, int cols) {
    int i = blockIdx.x * blockDim.x + threadIdx.x;
    if (i < n) out[i] = fmaxf(in[i] + bias[i % cols], 0.0f);
}

// mean / log_var heads: grid = (LAT/16, B/16, 2)
__global__ void heads_kernel(const float* __restrict__ he,
                             const float* __restrict__ Wm, const float* __restrict__ bm,
                             const float* __restrict__ Wl, const float* __restrict__ bl,
                             float* __restrict__ out_mean, float* __restrict__ out_lv) {
    int lane = threadIdx.x;
    int n0 = blockIdx.x * 16;
    int m0 = blockIdx.y * 16;
    const float* W  = blockIdx.z ? Wl : Wm;
    const float* bb = blockIdx.z ? bl : bm;
    float* out      = blockIdx.z ? out_lv : out_mean;
    v8f acc = V8F_ZERO;
    for (int k0 = 0; k0 < HID; k0 += 4) {
        v2f a = load_a_f32(he, HID, m0, k0, lane);
        v2f b = load_b_f32(W, LAT, k0, n0, lane);
        acc = wmma4(a, b, acc);
    }
    int n_lo = lane & 15;
    int mo   = (lane >> 4) * 8;
    float bc = bb[n0 + n_lo];
#pragma unroll
    for (int j = 0; j < 8; ++j) {
        int r = m0 + j + mo;
        out[(long)r * LAT + n0 + n_lo] = acc[j] + bc;
    }
}

__global__ void reparam_kernel(const float* __restrict__ mean, const float* __restrict__ lv,
                               const float* __restrict__ eps, float* __restrict__ z) {
    int i = blockIdx.x * blockDim.x + threadIdx.x;
    if (i < B_GR * LAT) z[i] = mean[i] + expf(0.5f * lv[i]) * eps[i];
}

// hd = relu(z @ W_d1 + b_d1): grid = (HID/16, B/16)
__global__ void dec1_kernel(const float* __restrict__ z, const float* __restrict__ Wd1,
                            const float* __restrict__ bd1, float* __restrict__ hd) {
    int lane = threadIdx.x;
    int n0 = blockIdx.x * 16;
    int m0 = blockIdx.y * 16;
    v8f acc = V8F_ZERO;
    for (int k0 = 0; k0 < LAT; k0 += 4) {
        v2f a = load_a_f32(z, LAT, m0, k0, lane);
        v2f b = load_b_f32(Wd1, HID, k0, n0, lane);
        acc = wmma4(a, b, acc);
    }
    int n_lo = lane & 15;
    int mo   = (lane >> 4) * 8;
    float bc = bd1[n0 + n_lo];
#pragma unroll
    for (int j = 0; j < 8; ++j) {
        int r = m0 + j + mo;
        hd[(long)r * HID + n0 + n_lo] = fmaxf(acc[j] + bc, 0.0f);
    }
}

// x_hat = sigmoid(hd @ W_d2 + b_d2): grid = IN_DIM/16 N-tiles, 1 wave/block,
// W_d2 streamed via async-LDS double buffering, B-tile reused across 4 M-tiles
__global__ void dec2_kernel(const float* __restrict__ hd, const float* __restrict__ Wd2,
                            const float* __restrict__ bd2, float* __restrict__ xhat) {
    __shared__ float lb[2][512];
    int lane = threadIdx.x;
    long n0 = (long)blockIdx.x * 16;
    v8f acc[4] = {V8F_ZERO, V8F_ZERO, V8F_ZERO, V8F_ZERO};

    stage_w(Wd2, IN_DIM, 0, n0, lb[0], lane);
    for (int c = 0; c < 15; ++c) {
        stage_w(Wd2, IN_DIM, (c + 1) * 32, n0, lb[(c + 1) & 1], lane);
        ASYNC_WAIT(4);
        mma_chunk(acc, lb[c & 1], hd, HID, c * 32, lane);
    }
    ASYNC_WAIT(0);
    mma_chunk(acc, lb[1], hd, HID, 15 * 32, lane);

    int n_lo = lane & 15;
    int mo   = (lane >> 4) * 8;
    float bc = bd2[n0 + n_lo];
#pragma unroll
    for (int t = 0; t < 4; ++t)
#pragma unroll
        for (int j = 0; j < 8; ++j) {
            int r = t * 16 + j + mo;
            float v = acc[t][j] + bc;
            xhat[(long)r * IN_DIM + n0 + n_lo] = 1.0f / (1.0f + expf(-v));
        }
}

extern "C" void kernel_launch(void* const* d_in, const int* in_sizes, int n_in,
                              void* d_out, int out_size, void* d_ws, size_t ws_size,
                              hipStream_t stream) {
    const float* x     = (const float*)d_in[0];
    const int*   ei    = (const int*)  d_in[1];
    const float* eps   = (const float*)d_in[2];
    const float* W_msg = (const float*)d_in[3];
    const float* b_msg = (const float*)d_in[4];
    const float* W_e1  = (const float*)d_in[5];
    const float* b_e1  = (const float*)d_in[6];
    const float* W_mean= (const float*)d_in[7];
    const float* b_mean= (const float*)d_in[8];
    const float* W_lv  = (const float*)d_in[9];
    const float* b_lv  = (const float*)d_in[10];
    const float* W_d1  = (const float*)d_in[11];
    const float* b_d1  = (const float*)d_in[12];
    const float* W_d2  = (const float*)d_in[13];
    const float* b_d2  = (const float*)d_in[14];

    float* out  = (float*)d_out;
    float* ws   = (float*)d_ws;
    float* s    = ws + WS_S;
    float* xs   = ws + WS_XS;
    float* deg  = ws + WS_DEG;
    float* pre  = ws + WS_PRE;
    float* he   = ws + WS_HE;
    float* z    = ws + WS_Z;
    float* hd   = ws + WS_HD;

    float* xhat = out;                              // [64, 64000]
    float* mean = out + (long)B_GR * IN_DIM;        // [64, 128]
    float* lv   = mean + (long)B_GR * LAT;          // [64, 128]

    const int* src = ei;
    const int* dst = ei + E_NUM;

    long nz = WS_HE;  // zero s + xs + deg + preact
    zero_kernel<<<(int)((nz + 255) / 256), 256, 0, stream>>>(ws, nz);
    deg_kernel<<<(E_NUM + 255) / 256, 256, 0, stream>>>(dst, deg);
    scatter_kernel<<<dim3(E_NUM / 4, B_GR), 256, 0, stream>>>(x, src, dst, s);
    conv_kernel<<<ROWS / 16, 32, 0, stream>>>(s, x, deg, W_msg, b_msg, xs);
    enc1_kernel<<<dim3(HID / 16, IN_DIM / 512), 32, 0, stream>>>(xs, W_e1, pre);
    bias_relu_kernel<<<(B_GR * HID + 255) / 256, 256, 0, stream>>>(pre, b_e1, he,
                                                                   B_GR * HID, HID);
    heads_kernel<<<dim3(LAT / 16, B_GR / 16, 2), 32, 0, stream>>>(he, W_mean, b_mean,
                                                                  W_lv, b_lv, mean, lv);
    reparam_kernel<<<(B_GR * LAT + 255) / 256, 256, 0, stream>>>(mean, lv, eps, z);
    dec1_kernel<<<dim3(HID / 16, B_GR / 16), 32, 0, stream>>>(z, W_d1, b_d1, hd);
    dec2_kernel<<<IN_DIM / 16, 32, 0, stream>>>(hd, W_d2, b_d2, xhat);
}